// GBST_40656160424320
// MI455X (gfx1250) — compile-verified
//
#include <hip/hip_runtime.h>

typedef float v2f __attribute__((ext_vector_type(2)));
typedef float v4f __attribute__((ext_vector_type(4)));
typedef float v8f __attribute__((ext_vector_type(8)));

#define DMODEL   128
#define HSTRIDE  132   // 128 + 4 pad -> conflict-free LDS transpose
#define TILE_L   48    // LCM(12,16): all block sizes m=1..4 align, 3 WMMA col-tiles

// intra-wave LDS producer->consumer fence (no cross-wave deps, so no s_barrier)
#define LDS_FENCE()                                          \
  do {                                                       \
    __builtin_amdgcn_wave_barrier();                         \
    asm volatile("s_wait_dscnt 0" ::: "memory");             \
    __builtin_amdgcn_wave_barrier();                         \
  } while (0)

__global__ __launch_bounds__(64) void gbst_fused(
    const float* __restrict__ x,        // (16, L)
    const float* __restrict__ conv_w,   // (128, 16, 4) == A (128x64) row-major
    const float* __restrict__ conv_b,   // (128)
    const float* __restrict__ score_w,  // (128)
    float* __restrict__ out_char,       // (L, 128)
    float* __restrict__ out_down,       // (ceil(L/4), 128)
    int L)
{
  __shared__ __align__(16) float lds_sw[DMODEL];
  __shared__ __align__(16) float lds_h[2][TILE_L * HSTRIDE];
  __shared__ __align__(16) float lds_sig[2][TILE_L];
  __shared__ __align__(16) float lds_p[2][TILE_L][4];

  const int wave = threadIdx.x >> 5;
  const int lane = threadIdx.x & 31;
  const int lh   = lane & 15;
  const bool lo16 = (lane < 16);

  // stage score_w once (before any divergence)
  if (threadIdx.x < 32)
    *(v4f*)&lds_sw[lane * 4] = *(const v4f*)&score_w[lane * 4];
  __syncthreads();

  const long tilebase = ((long)blockIdx.x * 2 + wave) * TILE_L;
  if (tilebase >= (long)L) return;   // wave-uniform

  // ---------------- Phase 1: conv via V_WMMA_F32_16X16X4_F32 ----------------
  // h[d,l] = sum_{i,k} w[d,i,k] * x[i, l+k-1]  (SAME pad: left 1, right 2)
  for (int ct = 0; ct < 3; ++ct) {
    const long l0 = tilebase + (long)ct * 16;
    // B fragment position: lanes 0-15 hold K rows {0,1}, lanes 16-31 rows {2,3}
    const long p  = l0 + (lo16 ? (long)(lh - 1) : (long)(lh + 1));
    const bool fast = (l0 >= 1) && (l0 + 18 <= (long)L);   // uniform

    v8f acc[8];
#pragma unroll
    for (int dt = 0; dt < 8; ++dt) {          // init C with bias
      const int dbase = dt * 16 + (lo16 ? 0 : 8);
      const v4f b0 = *(const v4f*)&conv_b[dbase];
      const v4f b1 = *(const v4f*)&conv_b[dbase + 4];
      acc[dt][0] = b0.x; acc[dt][1] = b0.y; acc[dt][2] = b0.z; acc[dt][3] = b0.w;
      acc[dt][4] = b1.x; acc[dt][5] = b1.y; acc[dt][6] = b1.z; acc[dt][7] = b1.w;
    }

#pragma unroll
    for (int s = 0; s < 16; ++s) {            // K-step = input channel s, taps k=0..3
      const float* xs = x + (size_t)s * (size_t)L;
      v2f bfrag;
      if (fast) {
        bfrag.x = xs[p];
        bfrag.y = xs[p + 1];
      } else {
        bfrag.x = (p >= 0     && p     < (long)L) ? xs[p]     : 0.0f;
        bfrag.y = (p + 1 >= 0 && p + 1 < (long)L) ? xs[p + 1] : 0.0f;
      }
      const int kap = 4 * s + (lo16 ? 0 : 2);
#pragma unroll
      for (int dt = 0; dt < 8; ++dt) {
        const int d = dt * 16 + lh;
        const v2f afrag = *(const v2f*)&conv_w[d * 64 + kap];  // 8B-aligned
        acc[dt] = __builtin_amdgcn_wmma_f32_16x16x4_f32(
            false, afrag, false, bfrag, (short)0, acc[dt], false, false);
      }
    }

    // transpose-store h tile to LDS as [l][d], zeroing columns l >= L
    const long gcol = l0 + lh;
    const float mask = (gcol < (long)L) ? 1.0f : 0.0f;
    const int row = ct * 16 + lh;
#pragma unroll
    for (int dt = 0; dt < 8; ++dt) {
      const int fb = row * HSTRIDE + dt * 16 + (lo16 ? 0 : 8);
      v4f lo, hi;
      lo.x = acc[dt][0] * mask; lo.y = acc[dt][1] * mask;
      lo.z = acc[dt][2] * mask; lo.w = acc[dt][3] * mask;
      hi.x = acc[dt][4] * mask; hi.y = acc[dt][5] * mask;
      hi.z = acc[dt][6] * mask; hi.w = acc[dt][7] * mask;
      *(v4f*)&lds_h[wave][fb]     = lo;
      *(v4f*)&lds_h[wave][fb + 4] = hi;
    }
  }
  LDS_FENCE();

  // ---------------- Phase 2: sigma(l) = score_w . h[l] ----------------
#pragma unroll
  for (int rep = 0; rep < 2; ++rep) {
    const int l = rep * 32 + lane;
    if (l < TILE_L) {
      const float* hr = &lds_h[wave][l * HSTRIDE];
      float sig = 0.0f;
#pragma unroll 8
      for (int d = 0; d < DMODEL; d += 4) {
        const v4f hv = *(const v4f*)&hr[d];
        const v4f sv = *(const v4f*)&lds_sw[d];
        sig += hv.x * sv.x + hv.y * sv.y + hv.z * sv.z + hv.w * sv.w;
      }
      lds_sig[wave][l] = sig;
    }
  }
  LDS_FENCE();

  // ---------------- Phase 3: softmax over block sizes; q_m = p_m / m ----------------
  // (score_b adds the same constant to all 4 logits -> cancels in softmax)
#pragma unroll
  for (int rep = 0; rep < 2; ++rep) {
    const int l = rep * 32 + lane;
    if (l < TILE_L) {
      const float* sg = lds_sig[wave];
      const int a2 = l & ~1;
      const int a3 = l - (l % 3);      // tilebase % 12 == 0 -> local alignment valid
      const int a4 = l & ~3;
      const float s1 = sg[l];
      const float s2 = (sg[a2] + sg[a2 + 1]) * 0.5f;
      const float s3 = (sg[a3] + sg[a3 + 1] + sg[a3 + 2]) * (1.0f / 3.0f);
      const float s4 = (sg[a4] + sg[a4 + 1] + sg[a4 + 2] + sg[a4 + 3]) * 0.25f;
      const float mx = fmaxf(fmaxf(s1, s2), fmaxf(s3, s4));
      const float e1 = expf(s1 - mx), e2 = expf(s2 - mx);
      const float e3 = expf(s3 - mx), e4 = expf(s4 - mx);
      const float inv = 1.0f / (e1 + e2 + e3 + e4);
      v4f q;
      q.x = e1 * inv;
      q.y = e2 * inv * 0.5f;
      q.z = e3 * inv * (1.0f / 3.0f);
      q.w = e4 * inv * 0.25f;
      *(v4f*)&lds_p[wave][l][0] = q;
    }
  }
  LDS_FENCE();

  // ---------------- Phase 4: out[l] = sum_m p_m * blockmean_m(h); fused down ----------------
  // Per 12-segment: out row = constant-sparsity 12x12 mix of h rows (folds at compile time).
#pragma unroll
  for (int seg = 0; seg < 4; ++seg) {
    v4f hreg[12];
#pragma unroll
    for (int op = 0; op < 12; ++op)
      hreg[op] = *(const v4f*)&lds_h[wave][(seg * 12 + op) * HSTRIDE + lane * 4];

    v4f dacc; dacc.x = dacc.y = dacc.z = dacc.w = 0.0f;
#pragma unroll
    for (int o = 0; o < 12; ++o) {
      const int l = seg * 12 + o;
      const long gl = tilebase + l;
      const v4f q = *(const v4f*)&lds_p[wave][l][0];   // broadcast read
      v4f a; a.x = a.y = a.z = a.w = 0.0f;
#pragma unroll
      for (int op = 0; op < 12; ++op) {
        // all conditions are compile-time constants after unrolling
        const bool any = (o == op) || ((o >> 1) == (op >> 1)) ||
                         ((o / 3) == (op / 3)) || ((o >> 2) == (op >> 2));
        if (any) {
          float w = 0.0f;
          if (o == op)               w += q.x;
          if ((o >> 1) == (op >> 1)) w += q.y;
          if ((o / 3)  == (op / 3))  w += q.z;
          if ((o >> 2) == (op >> 2)) w += q.w;
          a.x = fmaf(w, hreg[op].x, a.x);
          a.y = fmaf(w, hreg[op].y, a.y);
          a.z = fmaf(w, hreg[op].z, a.z);
          a.w = fmaf(w, hreg[op].w, a.w);
        }
      }
      dacc.x += a.x; dacc.y += a.y; dacc.z += a.z; dacc.w += a.w;

      if (gl < (long)L)
        *(v4f*)&out_char[(size_t)gl * DMODEL + lane * 4] = a;   // coalesced b128

      if ((o & 3) == 3) {
        if (gl - 3 < (long)L) {
          v4f dv;
          dv.x = dacc.x * 0.25f; dv.y = dacc.y * 0.25f;
          dv.z = dacc.z * 0.25f; dv.w = dacc.w * 0.25f;
          *(v4f*)&out_down[(size_t)(gl >> 2) * DMODEL + lane * 4] = dv;
        }
        dacc.x = dacc.y = dacc.z = dacc.w = 0.0f;
      }
    }
  }
}

extern "C" void kernel_launch(void* const* d_in, const int* in_sizes, int n_in,
                              void* d_out, int out_size, void* d_ws, size_t ws_size,
                              hipStream_t stream) {
  const float* x       = (const float*)d_in[0];  // (16, L)
  const float* conv_w  = (const float*)d_in[1];  // (128,16,4)
  const float* conv_b  = (const float*)d_in[2];  // (128)
  const float* score_w = (const float*)d_in[3];  // (128)
  // d_in[4] = score_b: cancels in softmax, unused.
  const int L = in_sizes[0] / 16;

  float* out_char = (float*)d_out;
  float* out_down = out_char + (size_t)L * DMODEL;

  const int nT  = (L + TILE_L - 1) / TILE_L;
  const int nWG = (nT + 1) / 2;                  // 2 waves (tiles) per workgroup
  gbst_fused<<<nWG, 64, 0, stream>>>(x, conv_w, conv_b, score_w,
                                     out_char, out_down, L);
}